// FactorVAE_Loss_5222680231998
// MI455X (gfx1250) — compile-verified
//
#include <hip/hip_runtime.h>
#include <math.h>

typedef __attribute__((ext_vector_type(2))) float v2f;
typedef __attribute__((ext_vector_type(8))) float v8f;

#define TWO_PI 6.283185307179586f
#define EPS    1e-6f

// Wave32 all-lane sum using V_WMMA_F32_16X16X4_F32 with B = ones.
// A layout (32-bit A 16x4): lanes 0-15 hold A[m][0..1] (vgpr0/1), lanes 16-31
// hold A[m][2..3]. We place the per-lane partial in component 0 and zero in
// component 1, so row m of A sums to p[lane m] + p[lane m+16].
// D layout: vgpr j holds D[j][n] for lanes 0-15 and D[j+8][n] for lanes 16-31,
// identical across columns n. Summing the 8 D vgprs gives rows 0-7 total in the
// low half-wave and rows 8-15 total in the high half-wave; one xor-16 shuffle
// merges them. EXEC is all-ones at every call site (no divergence).
__device__ __forceinline__ float wave_reduce_sum(float p) {
    v2f a; a.x = p;    a.y = 0.0f;
    v2f b; b.x = 1.0f; b.y = 1.0f;
    v8f c = {};
    v8f d = __builtin_amdgcn_wmma_f32_16x16x4_f32(
        /*neg_a=*/false, a, /*neg_b=*/false, b,
        /*c_mod=*/(short)0, c, /*reuse_a=*/false, /*reuse_b=*/false);
    float s = ((d[0] + d[1]) + (d[2] + d[3])) + ((d[4] + d[5]) + (d[6] + d[7]));
    s += __shfl_xor(s, 16, 32);
    return s;   // full 32-lane sum, uniform across the wave
}

__device__ __forceinline__ float logloss_term(float y, float mu, float s) {
    float diff = y - mu;
    float r = __builtin_amdgcn_rcpf(s);          // v_rcp_f32
    float q = diff * r;
    return 0.5f * q * q + 0.5f * __logf(TWO_PI * s);
}

// Pass 1: bandwidth-bound grid-stride reduction over B*T elements with
// float4 (global_load_b128) streams; one partial per block.
__global__ void __launch_bounds__(256)
logloss_partial_kernel(const float4* __restrict__ y,
                       const float4* __restrict__ ymu,
                       const float4* __restrict__ ysg,
                       float* __restrict__ partials,
                       int n4) {
    __shared__ float smem[8];
    const int tid    = threadIdx.x;
    const int stride = gridDim.x * blockDim.x;
    float acc = 0.0f;
    for (int i = blockIdx.x * blockDim.x + tid; i < n4; i += stride) {
        float4 yv = y[i];
        float4 mv = ymu[i];
        float4 sv = ysg[i];
        acc += logloss_term(yv.x, mv.x, sv.x);
        acc += logloss_term(yv.y, mv.y, sv.y);
        acc += logloss_term(yv.z, mv.z, sv.z);
        acc += logloss_term(yv.w, mv.w, sv.w);
    }
    float ws = wave_reduce_sum(acc);
    if ((tid & 31) == 0) smem[tid >> 5] = ws;
    __syncthreads();
    if (tid == 0) {
        float blk = 0.0f;
        #pragma unroll
        for (int w = 0; w < 8; ++w) blk += smem[w];
        partials[blockIdx.x] = blk;
    }
}

// Pass 2 (single block): reduce the block partials, compute the tiny diagonal
// KL reduction over K elements, combine into the final scalar. Deterministic:
// fixed traversal order, no atomics.
__global__ void __launch_bounds__(256)
final_reduce_kernel(const float* __restrict__ partials, int n_part,
                    const float* __restrict__ mu1, const float* __restrict__ s1,
                    const float* __restrict__ mu2, const float* __restrict__ s2,
                    int K, float inv_n, float* __restrict__ out) {
    __shared__ float smem[8];
    const int tid = threadIdx.x;

    float acc_log = 0.0f;
    for (int i = tid; i < n_part; i += blockDim.x) acc_log += partials[i];

    float acc_kl = 0.0f;
    for (int k = tid; k < K; k += blockDim.x) {
        float a  = fmaxf(s1[k], EPS);
        float b  = fmaxf(s2[k], EPS);
        float d  = mu1[k] - mu2[k];
        float rb = __builtin_amdgcn_rcpf(b);
        acc_kl += (a + d * d) * rb + __logf(b) - __logf(a);
    }

    float sum_log = wave_reduce_sum(acc_log);
    if ((tid & 31) == 0) smem[tid >> 5] = sum_log;
    __syncthreads();
    if (tid == 0) {
        float t = 0.0f;
        #pragma unroll
        for (int w = 0; w < 8; ++w) t += smem[w];
        smem[0] = t;                       // stash total logloss sum
    }
    __syncthreads();
    float total_log = smem[0];
    __syncthreads();

    float sum_kl = wave_reduce_sum(acc_kl);
    if ((tid & 31) == 0) smem[tid >> 5] = sum_kl;
    __syncthreads();
    if (tid == 0) {
        float t = 0.0f;
        #pragma unroll
        for (int w = 0; w < 8; ++w) t += smem[w];
        out[0] = total_log * inv_n + 0.5f * (t - (float)K);
    }
}

extern "C" void kernel_launch(void* const* d_in, const int* in_sizes, int n_in,
                              void* d_out, int out_size, void* d_ws, size_t ws_size,
                              hipStream_t stream) {
    const float* y    = (const float*)d_in[0];
    const float* ymu  = (const float*)d_in[1];
    const float* ysg  = (const float*)d_in[2];
    const float* mu1  = (const float*)d_in[3];
    const float* s1   = (const float*)d_in[4];
    const float* mu2  = (const float*)d_in[5];
    const float* s2   = (const float*)d_in[6];
    float* out = (float*)d_out;

    const int n  = in_sizes[0];        // B*T = 16,777,216 (multiple of 4)
    const int n4 = n / 4;
    const int K  = in_sizes[3];        // 8192

    const int block  = 256;
    int grid = 2048;                   // 8 f4-iterations/thread; fills all WGPs
    if (grid > (n4 + block - 1) / block) grid = (n4 + block - 1) / block;

    float* partials = (float*)d_ws;    // grid floats (8 KB) of scratch

    logloss_partial_kernel<<<grid, block, 0, stream>>>(
        (const float4*)y, (const float4*)ymu, (const float4*)ysg, partials, n4);

    final_reduce_kernel<<<1, block, 0, stream>>>(
        partials, grid, mu1, s1, mu2, s2, K, 1.0f / (float)n, out);
}